// CRF_6700148982294
// MI455X (gfx1250) — compile-verified
//
#include <hip/hip_runtime.h>
#include <math.h>

#define B_ 256
#define S_ 2048
#define L_ 64
#define START_ 62
#define END_ 63

#define LOG2E_ 1.4426950408889634f
#define LN2_   0.6931471805599453f

typedef float v2f __attribute__((ext_vector_type(2)));
typedef float v8f __attribute__((ext_vector_type(8)));

// Raw CDNA5 transcendentals (v_log_f32 = log2, v_exp_f32 = exp2).
// Trailing v_nop satisfies the TRANS-op co-execution hazard (ISA 7.4).
__device__ __forceinline__ float fast_log2(float x) {
    float r;
    asm("v_log_f32 %0, %1\n\tv_nop" : "=v"(r) : "v"(x));
    return r;
}
__device__ __forceinline__ float fast_exp2(float x) {
    float r;
    asm("v_exp_f32 %0, %1\n\tv_nop" : "=v"(r) : "v"(x));
    return r;
}

// Workspace layout (floats):
//   [0, 4096)    expTn[64][64]  : exp(T[j][k] - Tmax[j])
//   [4096, 4160) Tmax[64]
//   [4160, 4416) gold[256]

// ---------------------------------------------------------------------------
// Kernel 1: row-normalized exp of the transition matrix. 1 block x 64 threads.
// ---------------------------------------------------------------------------
__global__ void crf_prep(const float* __restrict__ T, float* __restrict__ ws) {
    int j = threadIdx.x;               // 0..63, one row per thread
    float m = -INFINITY;
    for (int k = 0; k < L_; ++k) m = fmaxf(m, T[j * L_ + k]);
    ws[4096 + j] = m;
    for (int k = 0; k < L_; ++k) ws[j * L_ + k] = __expf(T[j * L_ + k] - m);
}

// ---------------------------------------------------------------------------
// Kernel 2: gold path score. One block per batch, 256 threads.
// ---------------------------------------------------------------------------
__global__ __launch_bounds__(256) void crf_gold(const float* __restrict__ logits,
                                                const float* __restrict__ T,
                                                const int* __restrict__ labels,
                                                const int* __restrict__ lens,
                                                float* __restrict__ ws) {
    const int b = blockIdx.x;
    const int tid = threadIdx.x;
    __shared__ float Ts[L_ * L_];
    for (int i = tid; i < L_ * L_; i += 256) Ts[i] = T[i];
    __syncthreads();

    const int len = lens[b];
    const int* lab = labels + (size_t)b * S_;
    const float* lg = logits + (size_t)b * S_ * L_;

    float acc = 0.0f;
    // term(i) = T[e(i+1), e(i)] for i in [0, len]; unary for i in [0, len)
    for (int i = tid; i <= len; i += 256) {
        int prev = (i == 0) ? START_ : lab[i - 1];
        int cur  = (i == len) ? END_ : lab[i];
        acc += Ts[cur * L_ + prev];
        if (i < len) acc += lg[(size_t)i * L_ + lab[i]];
    }

    __shared__ float red[256];
    red[tid] = acc;
    __syncthreads();
    for (int s = 128; s > 0; s >>= 1) {
        if (tid < s) red[tid] += red[tid + s];
        __syncthreads();
    }
    if (tid == 0) ws[4160 + b] = red[0];
}

// ---------------------------------------------------------------------------
// Kernel 3: forward recursion via f32 WMMA. 16 blocks (16 batches each),
// 128 threads = 4 wave32. Wave w owns j-rows [16w, 16w+16).
// P is stored in B-fragment-pair layout: element (k,b) lives at
//   p_lds[(k>>1)*32 + 2*b + (k&1)]
// so each B fragment is one aligned ds_load_b64 (conflict-free across the
// wave). All renormalization bookkeeping is kept in the log2 domain so
// v_log_f32/v_exp_f32 are used raw; converted by ln2 once at the end.
// ---------------------------------------------------------------------------
__global__ __launch_bounds__(128) void crf_fwd(const float* __restrict__ logits,
                                               const int* __restrict__ lens,
                                               const float* __restrict__ ws,
                                               float* __restrict__ out) {
    const int tid  = threadIdx.x;
    const int wave = tid >> 5;
    const int lane = tid & 31;
    const int half = lane >> 4;    // 0: lanes 0-15, 1: lanes 16-31
    const int col  = lane & 15;    // batch column within tile
    const int jbase = wave * 16;
    const int b0 = blockIdx.x * 16;

    const float* expTn = ws;            // [64][64]
    const float* Tmax  = ws + 4096;     // [64]
    const float* gold  = ws + 4160;     // [256]

    __shared__ float p_lds[32 * 32];    // (k-pair) x (16 cols x 2) = 4KB
    __shared__ float maxbuf[64];        // per-wave combined column maxima
    __shared__ float off2[16];          // per-column offset, log2 units

    // A fragments in registers for the whole loop.
    // 32-bit A(16x4) layout: M = lane%16, K = 4*kc + 2*half + v.
    v2f afrag[16];
    {
        const float* arow = expTn + (jbase + col) * L_;
#pragma unroll
        for (int kc = 0; kc < 16; ++kc) {
            afrag[kc].x = arow[kc * 4 + half * 2 + 0];
            afrag[kc].y = arow[kc * 4 + half * 2 + 1];
        }
    }
    // Tmax (in log2 units) for the D rows this lane owns: j = jbase+8*half+r.
    float tmr2[8];
#pragma unroll
    for (int r = 0; r < 8; ++r) tmr2[r] = Tmax[jbase + half * 8 + r] * LOG2E_;

    const int lenb = lens[b0 + col];

    // alpha0 = -100 except 0 at START; off = 0; p = exp(alpha0 - 0)
    const float p_cold = __expf(-100.0f);
    for (int i = tid; i < L_ * 16; i += 128) {
        int j = i >> 4, b = i & 15;
        p_lds[(j >> 1) * 32 + 2 * b + (j & 1)] = (j == START_) ? 1.0f : p_cold;
    }
    if (tid < 16) off2[tid] = 0.0f;
    __syncthreads();

    const float* lgbase = logits + ((size_t)(b0 + col) * S_) * L_ + jbase + half * 8;

    for (int t = 0; t < S_; ++t) {
        // This lane's 8 logits (contiguous j), 32B aligned.
        const float* lp = lgbase + (size_t)t * L_;
        float4 lo0 = *(const float4*)(lp);
        float4 lo1 = *(const float4*)(lp + 4);
        if (t + 1 < S_) __builtin_prefetch(lp + L_, 0, 3);

        // All 16 B fragments as one DS clause: pair (K, K+1), K = 4*kc+2*half.
        v2f bf[16];
#pragma unroll
        for (int kc = 0; kc < 16; ++kc)
            bf[kc] = *(const v2f*)&p_lds[(2 * kc + half) * 32 + 2 * col];

        // Two independent accumulation chains, merged at the end.
        v8f c0 = {}, c1 = {};
#pragma unroll
        for (int kc = 0; kc < 16; kc += 2) {
            c0 = __builtin_amdgcn_wmma_f32_16x16x4_f32(
                false, afrag[kc], false, bf[kc], (short)0, c0, false, false);
            c1 = __builtin_amdgcn_wmma_f32_16x16x4_f32(
                false, afrag[kc + 1], false, bf[kc + 1], (short)0, c1, false, false);
        }
        v8f c = c0 + c1;

        // q2[j,b] = (logit + Tmax[j]) * log2e + log2(D); column max over 64 j.
        float q2[8];
        float lm = -INFINITY;
#pragma unroll
        for (int r = 0; r < 8; ++r) {
            float l = (r < 4) ? ((const float*)&lo0)[r] : ((const float*)&lo1)[r - 4];
            q2[r] = fmaf(l, LOG2E_, tmr2[r]) + fast_log2(c[r]);
            lm = fmaxf(lm, q2[r]);
        }
        // Combine the two half-rows within the wave, then exchange across waves.
        lm = fmaxf(lm, __shfl_xor(lm, 16, 32));
        if (half == 0) maxbuf[wave * 16 + col] = lm;
        __syncthreads();   // also orders this step's P reads before P writes

        float m2 = fmaxf(fmaxf(maxbuf[col], maxbuf[16 + col]),
                         fmaxf(maxbuf[32 + col], maxbuf[48 + col]));

        const bool act = (t < lenb);
        if (act) {
#pragma unroll
            for (int r = 0; r < 8; r += 2) {
                const int j = jbase + half * 8 + r;   // even
                v2f pr;
                pr.x = fast_exp2(q2[r] - m2);
                pr.y = fast_exp2(q2[r + 1] - m2);
                *(v2f*)&p_lds[(j >> 1) * 32 + 2 * col] = pr;
            }
            if (tid < 16) off2[col] += m2;   // wave0/half0: col == tid
        }
        __syncthreads();   // P writes visible before next step's reads
    }

    // norm[b] = off2[b]*ln2 + Tmax[END] + log(sum_j p[j,b] * expTn[END][j])
    if (tid < 16) {
        const int b = tid;
        const float* erow = expTn + END_ * L_;
        float s = 0.0f;
        for (int j = 0; j < L_; ++j)
            s += p_lds[(j >> 1) * 32 + 2 * b + (j & 1)] * erow[j];
        out[b0 + b] = gold[b0 + b] -
                      (off2[b] * LN2_ + Tmax[END_] + fast_log2(s) * LN2_);
    }
}

// ---------------------------------------------------------------------------
extern "C" void kernel_launch(void* const* d_in, const int* in_sizes, int n_in,
                              void* d_out, int out_size, void* d_ws, size_t ws_size,
                              hipStream_t stream) {
    const float* logits = (const float*)d_in[0];   // (256, 2048, 64) f32
    const float* trans  = (const float*)d_in[1];   // (64, 64) f32
    const int*   labels = (const int*)d_in[2];     // (256, 2048) i32
    const int*   lens   = (const int*)d_in[3];     // (256,) i32
    float* out = (float*)d_out;                    // (256,) f32
    float* ws  = (float*)d_ws;                     // >= 4416 floats

    crf_prep<<<1, 64, 0, stream>>>(trans, ws);
    crf_gold<<<B_, 256, 0, stream>>>(logits, trans, labels, lens, ws);
    crf_fwd<<<B_ / 16, 128, 0, stream>>>(logits, lens, ws, out);
}